// _UCELoss_17343077941753
// MI455X (gfx1250) — compile-verified
//
#include <hip/hip_runtime.h>
#include <hip/hip_bf16.h>

// UCE loss: 10-bin histogram reduction over 2^25 f32 pairs.
// Memory-bound: 256 MB read -> ~11us at 23.3 TB/s HBM. Strategy:
//   - float4 (b128) streaming loads, grid-stride
//   - per-thread private LDS histogram rows (stride 33 -> bank-conflict-free
//     for wave32), updated with ds_add_f32 atomics (no RMW dependency stall)
//   - wave-level reduction of the 30 partial sums via chained
//     v_wmma_f32_16x16x4_f32 with column-one-hot B (f32-exact)
//   - per-block partials to workspace, tiny deterministic finalize kernel

typedef __attribute__((ext_vector_type(2))) float v2f;
typedef __attribute__((ext_vector_type(8))) float v8f;

#define NBINS   10
#define BLOCK   256     // 8 waves of 32
#define GRID    1024
#define TSTRIDE 33      // per-thread LDS row stride (floats): conflict-free

// Reduce 15 per-lane f32 accumulators across the wave using WMMA.
// Chain D += A x B with A = (acc_j, 0) per lane and B = one-hot column j.
// Documented A layout (16x4 f32): lane L supplies A[L%16][2*(L/16) + vgpr].
// => D[m][j] accumulates acc_j(lane m) + acc_j(lane m+16).
// Documented D layout: lane L, vgpr v holds D[v + 8*(L/16)][L%16].
// Summing the 8 D vgprs then adding the cross-half value (shfl_xor 16)
// yields, on every lane, the full 32-lane total of accumulator (lane%16).
__device__ __forceinline__ float wave_reduce15_wmma(const float* acc, int col) {
    v8f d = {0.f, 0.f, 0.f, 0.f, 0.f, 0.f, 0.f, 0.f};
#pragma unroll
    for (int j = 0; j < 15; ++j) {
        v2f a;
        a.x = acc[j];
        a.y = 0.0f;
        float oh = (col == j) ? 1.0f : 0.0f;  // one-hot column select
        v2f b;
        b.x = oh;
        b.y = oh;
        d = __builtin_amdgcn_wmma_f32_16x16x4_f32(
                /*neg_a=*/false, a, /*neg_b=*/false, b,
                /*c_mod=*/(short)0, d, /*reuse_a=*/false, /*reuse_b=*/false);
    }
    float s = d[0] + d[1] + d[2] + d[3] + d[4] + d[5] + d[6] + d[7];
    // combine the two row-halves (rows 0..7 held by lanes 0-15, 8..15 by 16-31)
    s += __shfl_xor(s, 16);
    return s;  // total for accumulator (lane % 16), valid for col < 15
}

__global__ void __launch_bounds__(BLOCK)
uce_partial(const float* __restrict__ U, const float* __restrict__ E,
            float* __restrict__ partials, int n4) {
    __shared__ float hist[BLOCK * TSTRIDE];  // 33 KB: slots 0..29 stats, 30..32 trash
    __shared__ float blockAcc[32];

    const int tid = threadIdx.x;
    float* my = &hist[tid * TSTRIDE];
#pragma unroll
    for (int s = 0; s < TSTRIDE; ++s) my[s] = 0.0f;
    if (tid < 32) blockAcc[tid] = 0.0f;
    __syncthreads();

    const float4* U4 = (const float4*)U;
    const float4* E4 = (const float4*)E;
    const int stride = GRID * BLOCK;
    for (int i = blockIdx.x * BLOCK + tid; i < n4; i += stride) {
        float4 u4 = U4[i];
        float4 e4 = E4[i];
        float us[4] = {u4.x, u4.y, u4.z, u4.w};
        float es[4] = {e4.x, e4.y, e4.z, e4.w};
#pragma unroll
        for (int k = 0; k < 4; ++k) {
            float u = us[k];
            float e = es[k];
            // bin b iff b*0.1 < u <= (b+1)*0.1  (torch gt/le semantics)
            int b = (int)__builtin_ceilf(u * 10.0f) - 1;
            b = b < 0 ? 0 : (b > 9 ? 9 : b);
            float lo = (float)b * 0.1f;
            float hi = (float)(b + 1) * 0.1f;
            int bc = b + ((u > hi) ? 1 : 0) - ((b > 0 && u <= lo) ? 1 : 0);
            bc = bc < 0 ? 0 : (bc > 9 ? 9 : bc);
            bool valid = (u > 0.0f) && (u <= 1.0f);
            // flat layout: 0..9 = count, 10..19 = sum_u, 20..29 = sum_e
            atomicAdd(&my[valid ? bc : 30], 1.0f);
            atomicAdd(&my[valid ? (10 + bc) : 31], u);
            atomicAdd(&my[valid ? (20 + bc) : 32], e);
        }
    }

    // read back own-thread slots (same-wave LDS ops are ordered)
    float acc[30];
#pragma unroll
    for (int j = 0; j < 30; ++j) acc[j] = my[j];

    const int lane = tid & 31;
    const int col = lane & 15;
    // WMMA requires EXEC all-ones: straight-line, all threads execute
    float tot0 = wave_reduce15_wmma(&acc[0], col);    // stats 0..14
    float tot1 = wave_reduce15_wmma(&acc[15], col);   // stats 15..29

    if (lane < 15) {
        atomicAdd(&blockAcc[col], tot0);
        atomicAdd(&blockAcc[15 + col], tot1);
    }
    __syncthreads();
    if (tid < 30) partials[blockIdx.x * 32 + tid] = blockAcc[tid];
}

__global__ void uce_final(const float* __restrict__ partials,
                          float* __restrict__ out, int nblocks, float n) {
    __shared__ float sums[32];
    const int j = threadIdx.x;  // 32 threads, 1 wave
    float s = 0.0f;
    if (j < 30) {
        for (int b = 0; b < nblocks; ++b) s += partials[b * 32 + j];
    }
    sums[j] = s;
    __syncthreads();
    if (j == 0) {
        float uce = 0.0f;
        for (int b = 0; b < NBINS; ++b) {
            float c = sums[b];
            float su = sums[10 + b];
            float se = sums[20 + b];
            float denom = (c > 0.0f) ? c : 1.0f;
            float avg_u = su / denom;
            float avg_e = se / denom;
            float prop = c / n;
            if (c > 0.0f) uce += __builtin_fabsf(avg_u - avg_e) * prop;
        }
        out[0] = uce;
    }
}

extern "C" void kernel_launch(void* const* d_in, const int* in_sizes, int n_in,
                              void* d_out, int out_size, void* d_ws, size_t ws_size,
                              hipStream_t stream) {
    (void)n_in; (void)out_size; (void)ws_size;
    const float* U = (const float*)d_in[0];
    const float* E = (const float*)d_in[1];
    float* out = (float*)d_out;
    float* partials = (float*)d_ws;  // GRID * 32 floats = 128 KB

    const int n = in_sizes[0];
    const int n4 = n / 4;  // N = 2^25, divisible by 4

    uce_partial<<<GRID, BLOCK, 0, stream>>>(U, E, partials, n4);
    uce_final<<<1, 32, 0, stream>>>(partials, out, GRID, (float)n);
}